// MEBLoss_49280454754360
// MI455X (gfx1250) — compile-verified
//
#include <hip/hip_runtime.h>
#include <hip/hip_bf16.h>
#include <math.h>

typedef float v2f __attribute__((ext_vector_type(2)));
typedef float v8f __attribute__((ext_vector_type(8)));

#define TAU_B    0.5f
#define MARGIN_M 0.5f
#define IR_PER_BLOCK 8   // rows (waves) per block in intra kernel

__device__ __forceinline__ float waveReduceSum(float x) {
#pragma unroll
    for (int m = 16; m >= 1; m >>= 1)
        x += __shfl_xor(x, m, 32);
    return x;
}

// ---------------------------------------------------------------------------
// Kernel 1: center row squared-norms (padded to Mpad) + rf = |r| + 1e-6
// one wave32 per padded row
// ---------------------------------------------------------------------------
__global__ void norms_kernel(const float* __restrict__ centers,
                             const float* __restrict__ radii,
                             float* __restrict__ norms,
                             float* __restrict__ rf,
                             int M, int Mpad, int D) {
    int gtid = blockIdx.x * blockDim.x + threadIdx.x;
    int wave = gtid >> 5;
    int lane = threadIdx.x & 31;
    if (wave < Mpad) {
        float s = 0.f;
        if (wave < M) {
            const float* row = centers + (size_t)wave * D;
            for (int j = lane; j < D; j += 32) {
                float v = row[j];
                s = fmaf(v, v, s);
            }
        }
        s = waveReduceSum(s);
        if (lane == 0) norms[wave] = s;
    }
    if (gtid < Mpad) rf[gtid] = (gtid < M) ? (fabsf(radii[gtid]) + 1e-6f) : 0.f;
}

// ---------------------------------------------------------------------------
// Kernel 2: Gram tiles via V_WMMA_F32_16X16X4_F32; overlap + diversity partials
// grid = (Mpad/16)^2 blocks of exactly 32 threads (1 wave -> EXEC all ones)
// ---------------------------------------------------------------------------
__global__ void gram_kernel(const float* __restrict__ centers,
                            const float* __restrict__ norms,
                            const float* __restrict__ rf,
                            float* __restrict__ ovPart,
                            float* __restrict__ divPart,
                            int M, int D, int tilesPerRow) {
    const int tile = blockIdx.x;
    const int ti = tile % tilesPerRow;
    const int tj = tile / tilesPerRow;
    const int i0 = ti * 16, j0 = tj * 16;
    const int lane = threadIdx.x;

    // A tile rows (Gram rows), B tile rows (Gram cols); identical addressing
    const int rA = i0 + (lane & 15);
    const int rB = j0 + (lane & 15);
    const int khalf = (lane >> 4) << 1;              // 0 or 2 (K-pair select)
    const float mA = (rA < M) ? 1.f : 0.f;           // zero-pad without OOB
    const float mB = (rB < M) ? 1.f : 0.f;
    const int rAc = (rA < M) ? rA : (M - 1);
    const int rBc = (rB < M) ? rB : (M - 1);
    const float2* rowA = (const float2*)(centers + (size_t)rAc * D);
    const float2* rowB = (const float2*)(centers + (size_t)rBc * D);

    v8f acc = {0.f, 0.f, 0.f, 0.f, 0.f, 0.f, 0.f, 0.f};
#pragma unroll 4
    for (int k = 0; k < D; k += 4) {
        int kk = (k + khalf) >> 1;                   // float2 index
        float2 fa = rowA[kk];
        float2 fb = rowB[kk];
        v2f a; a.x = fa.x * mA; a.y = fa.y * mA;     // A[row][K], K=(lane/16)*2+{0,1}
        v2f b; b.x = fb.x * mB; b.y = fb.y * mB;     // B[K][col] = Cf[col][K]
        acc = __builtin_amdgcn_wmma_f32_16x16x4_f32(
            false, a, false, b, (short)0, acc, false, false);
    }

    // D layout: VGPR v, lanes 0-15 -> M=v, N=lane ; lanes 16-31 -> M=v+8
    const int n = j0 + (lane & 15);
    const int mbase = i0 + ((lane >> 4) << 3);
    float ov = 0.f, dv = 0.f;
#pragma unroll
    for (int v = 0; v < 8; ++v) {
        int m = mbase + v;
        if (m < M && n < M && m != n) {
            float dsq = norms[m] + norms[n] - 2.f * acc[v];
            float dd = sqrtf(fmaxf(dsq, 0.f));
            ov += fmaxf(rf[m] + rf[n] + MARGIN_M - dd, 0.f);
            // within-class pair p<q with K=2: (m even, n = m+1)
            if (n == m + 1 && (m & 1) == 0)
                dv += fmaxf(1.f - dd, 0.f);
        }
    }
    ov = waveReduceSum(ov);
    dv = waveReduceSum(dv);
    if (lane == 0) { ovPart[tile] = ov; divPart[tile] = dv; }
}

// ---------------------------------------------------------------------------
// Kernel 3: intra term — HBM-bound stream of z, one wave32 per row
// ---------------------------------------------------------------------------
__global__ void intra_kernel(const float* __restrict__ z,
                             const int* __restrict__ labels,
                             const float* __restrict__ centers,
                             const float* __restrict__ radii,
                             float* __restrict__ part,
                             int N, int D) {
    __shared__ float lds[IR_PER_BLOCK];
    const int wave = threadIdx.x >> 5;
    const int lane = threadIdx.x & 31;
    const int row = blockIdx.x * IR_PER_BLOCK + wave;

    float contrib = 0.f;
    if (row < N) {
        const int lab = labels[row];
        const float4* z4 = (const float4*)(z + (size_t)row * D);
        const float4* c0 = (const float4*)(centers + ((size_t)lab * 2 + 0) * D);
        const float4* c1 = (const float4*)(centers + ((size_t)lab * 2 + 1) * D);
        float s0 = 0.f, s1 = 0.f;
        const int nvec = D >> 2;                      // 64 float4 per row
        for (int idx = lane; idx < nvec; idx += 32) { // fully coalesced B128 loads
            float4 zz = z4[idx];
            float4 a = c0[idx];
            float4 b = c1[idx];
            float d;
            d = zz.x - a.x; s0 = fmaf(d, d, s0);
            d = zz.y - a.y; s0 = fmaf(d, d, s0);
            d = zz.z - a.z; s0 = fmaf(d, d, s0);
            d = zz.w - a.w; s0 = fmaf(d, d, s0);
            d = zz.x - b.x; s1 = fmaf(d, d, s1);
            d = zz.y - b.y; s1 = fmaf(d, d, s1);
            d = zz.z - b.z; s1 = fmaf(d, d, s1);
            d = zz.w - b.w; s1 = fmaf(d, d, s1);
        }
        s0 = waveReduceSum(s0);
        s1 = waveReduceSum(s1);
        if (lane == 0) {
            // exact 2-way softmax over (-dsq/tau)
            float t0 = -s0 / TAU_B, t1 = -s1 / TAU_B;
            float mx = fmaxf(t0, t1);
            float e0 = expf(t0 - mx), e1 = expf(t1 - mx);
            float inv = 1.f / (e0 + e1);
            float q0 = e0 * inv, q1 = e1 * inv;
            float dw = q0 * s0 + q1 * s1;
            float r0 = fabsf(radii[lab * 2 + 0]) + 1e-6f;
            float r1 = fabsf(radii[lab * 2 + 1]) + 1e-6f;
            float rw = q0 * r0 * r0 + q1 * r1 * r1;
            contrib = fmaxf(dw - rw, 0.f);           // ETA = 1
        }
    }
    if (lane == 0) lds[wave] = contrib;
    __syncthreads();
    if (threadIdx.x == 0) {
        float s = 0.f;
#pragma unroll
        for (int w = 0; w < IR_PER_BLOCK; ++w) s += lds[w];
        part[blockIdx.x] = s;
    }
}

// ---------------------------------------------------------------------------
// Kernel 4: deterministic final reduction + loss assembly
// ---------------------------------------------------------------------------
__global__ void finalize_kernel(const float* __restrict__ intraPart, int nIntra,
                                const float* __restrict__ ovPart,
                                const float* __restrict__ divPart, int nTiles,
                                float* __restrict__ out, int N, int M, int C) {
    __shared__ float lds[256];
    const int tid = threadIdx.x;

    float s = 0.f;
    for (int i = tid; i < nIntra; i += 256) s += intraPart[i];
    lds[tid] = s; __syncthreads();
    for (int off = 128; off >= 1; off >>= 1) {
        if (tid < off) lds[tid] += lds[tid + off];
        __syncthreads();
    }
    float intraSum = lds[0]; __syncthreads();

    s = 0.f;
    for (int i = tid; i < nTiles; i += 256) s += ovPart[i];
    lds[tid] = s; __syncthreads();
    for (int off = 128; off >= 1; off >>= 1) {
        if (tid < off) lds[tid] += lds[tid + off];
        __syncthreads();
    }
    float ovSum = lds[0]; __syncthreads();

    s = 0.f;
    for (int i = tid; i < nTiles; i += 256) s += divPart[i];
    lds[tid] = s; __syncthreads();
    for (int off = 128; off >= 1; off >>= 1) {
        if (tid < off) lds[tid] += lds[tid + off];
        __syncthreads();
    }
    float divSum = lds[0];

    if (tid == 0) {
        float Li = intraSum / (float)N;
        float Lo = ovSum / (float)(M * (M - 1));
        float Ld = divSum / (float)C;                 // C*K*(K-1)/2 == C for K=2
        float tot = Li + Lo + 0.5f * Ld;              // LAM_IN=1, LAM_OV=1, LAM_DIV=0.5
        out[0] = tot; out[1] = Li; out[2] = Lo; out[3] = Ld;
    }
}

// ---------------------------------------------------------------------------
extern "C" void kernel_launch(void* const* d_in, const int* in_sizes, int n_in,
                              void* d_out, int out_size, void* d_ws, size_t ws_size,
                              hipStream_t stream) {
    const float* z       = (const float*)d_in[0];
    const int*   labels  = (const int*)d_in[1];   // jax default: int64 -> int32
    const float* centers = (const float*)d_in[2];
    const float* radii   = (const float*)d_in[3];

    const int N  = in_sizes[1];
    const int D  = in_sizes[0] / N;               // 256
    const int M  = in_sizes[3];                   // C*K = 200
    const int K  = 2;
    const int C  = M / K;
    const int Mpad = ((M + 15) / 16) * 16;        // 208
    const int tpr = Mpad / 16;                    // 13
    const int nTiles = tpr * tpr;                 // 169
    const int nIntra = (N + IR_PER_BLOCK - 1) / IR_PER_BLOCK;

    float* ws        = (float*)d_ws;
    float* intraPart = ws;                        // [nIntra]
    float* ovPart    = intraPart + nIntra;        // [nTiles]
    float* divPart   = ovPart + nTiles;           // [nTiles]
    float* norms     = divPart + nTiles;          // [Mpad]
    float* rf        = norms + Mpad;              // [Mpad]

    int normBlocks = (Mpad * 32 + 255) / 256;
    norms_kernel<<<normBlocks, 256, 0, stream>>>(centers, radii, norms, rf, M, Mpad, D);
    gram_kernel<<<nTiles, 32, 0, stream>>>(centers, norms, rf, ovPart, divPart, M, D, tpr);
    intra_kernel<<<nIntra, 32 * IR_PER_BLOCK, 0, stream>>>(z, labels, centers, radii,
                                                           intraPart, N, D);
    finalize_kernel<<<1, 256, 0, stream>>>(intraPart, nIntra, ovPart, divPart, nTiles,
                                           (float*)d_out, N, M, C);
}